// GraphSAGELayer_74491912781907
// MI455X (gfx1250) — compile-verified
//
#include <hip/hip_runtime.h>

typedef float v2f __attribute__((ext_vector_type(2)));
typedef float v8f __attribute__((ext_vector_type(8)));

#define D_FEAT 64
#define D_CAT 128           // 2 * D_FEAT
#define LDS_STRIDE 130      // 128 + 2 pad -> conflict-free column access (130 % 64 == 2)

// ---------------------------------------------------------------------------
// Kernel 1: zero the workspace (accumulator [N*64] + degree count [N])
// ---------------------------------------------------------------------------
__global__ void sage_zero_kernel(float* __restrict__ ws, int n) {
    int i = blockIdx.x * blockDim.x + threadIdx.x;
    if (i < n) ws[i] = 0.0f;
}

// ---------------------------------------------------------------------------
// Kernel 2: edge scatter. One wave32 per edge; lane handles features
// {lane, lane+32}. Native global_atomic_add_f32 (L2-resident RMW).
// ---------------------------------------------------------------------------
__global__ void sage_scatter_kernel(const float* __restrict__ x,
                                    const int* __restrict__ row,
                                    const int* __restrict__ col,
                                    float* __restrict__ accum,
                                    float* __restrict__ cnt,
                                    int nEdges) {
    int gid = blockIdx.x * blockDim.x + threadIdx.x;
    int e = gid >> 5;
    int lane = gid & 31;
    if (e >= nEdges) return;
    int dst = row[e];   // segment id (aggregation target)
    int src = col[e];   // neighbour whose features are gathered
    const float* xs = x + (size_t)src * D_FEAT;
    float v0 = xs[lane];
    float v1 = xs[lane + 32];
    float* ad = accum + (size_t)dst * D_FEAT;
    unsafeAtomicAdd(ad + lane, v0);
    unsafeAtomicAdd(ad + lane + 32, v1);
    if (lane == 0) unsafeAtomicAdd(cnt + dst, 1.0f);
}

// ---------------------------------------------------------------------------
// Kernel 3: fused mean + concat + GEMM(+bias,ReLU) via V_WMMA_F32_16X16X4_F32.
// Block = 128 threads = 4 waves, one 16-node tile per block; wave w computes
// output columns [16w, 16w+16). K=128 walked in 32 fp32 WMMA steps.
// ---------------------------------------------------------------------------
__global__ void __launch_bounds__(128)
sage_gemm_kernel(const float* __restrict__ x,
                 const float* __restrict__ accum,
                 const float* __restrict__ cnt,
                 const float* __restrict__ W,     // [64, 128] row-major
                 const float* __restrict__ bias,  // [64]
                 float* __restrict__ out,         // [N, 64]
                 int nNodes) {
    __shared__ float sW[D_FEAT * LDS_STRIDE];   // 64 x 128 (+pad)  ~33 KB
    __shared__ float sC[16 * LDS_STRIDE];       // 16 x 128 (+pad)  ~8 KB

    const int tid = threadIdx.x;
    const int node0 = blockIdx.x * 16;

    // Stage W into LDS (read once per block; L2-resident across blocks).
    for (int idx = tid; idx < D_FEAT * D_CAT; idx += 128) {
        int o = idx >> 7;        // output feature (row of W)
        int k = idx & 127;       // input feature
        sW[o * LDS_STRIDE + k] = W[idx];
    }
    // Build combined = [x_i , neighbour_mean_i] for 16 nodes.
    for (int idx = tid; idx < 16 * D_CAT; idx += 128) {
        int m = idx >> 7;
        int f = idx & 127;
        int node = node0 + m;
        float v = 0.0f;
        if (node < nNodes) {
            if (f < D_FEAT) {
                v = x[(size_t)node * D_FEAT + f];
            } else {
                v = accum[(size_t)node * D_FEAT + (f - D_FEAT)]
                    / (cnt[node] + 1e-9f);
            }
        }
        sC[m * LDS_STRIDE + f] = v;
    }
    __syncthreads();

    const int lane = tid & 31;
    const int n0   = (tid >> 5) * 16;   // wave -> output column tile base
    const int mrow = lane & 15;         // A-fragment row  (M)
    const int ncol = lane & 15;         // B-fragment col  (N)
    const int kb   = (lane >> 4) << 1;  // lane-half K offset {0,2}

    v8f acc = {};
#pragma unroll
    for (int k0 = 0; k0 < D_CAT; k0 += 4) {
        v2f a, bb;
        // A 16x4 fp32: VGPR0 -> K = kb, VGPR1 -> K = kb+1
        a.x = sC[mrow * LDS_STRIDE + k0 + kb];
        a.y = sC[mrow * LDS_STRIDE + k0 + kb + 1];
        // B 4x16 fp32: B[k][n] = W[n][k]
        bb.x = sW[(n0 + ncol) * LDS_STRIDE + k0 + kb];
        bb.y = sW[(n0 + ncol) * LDS_STRIDE + k0 + kb + 1];
        acc = __builtin_amdgcn_wmma_f32_16x16x4_f32(
            /*neg_a=*/false, a, /*neg_b=*/false, bb,
            /*c_mod=*/(short)0, acc, /*reuse_a=*/false, /*reuse_b=*/false);
    }

    // Epilogue: C/D layout -> VGPR v, lane l: M = v + 8*(l>>4), N = l&15.
    const float bv = bias[n0 + ncol];
    const int mhi = (lane >> 4) << 3;
#pragma unroll
    for (int v = 0; v < 8; ++v) {
        int node = node0 + mhi + v;
        if (node < nNodes) {
            float r = acc[v] + bv;
            out[(size_t)node * D_FEAT + n0 + ncol] = fmaxf(r, 0.0f);
        }
    }
}

// ---------------------------------------------------------------------------
extern "C" void kernel_launch(void* const* d_in, const int* in_sizes, int n_in,
                              void* d_out, int out_size, void* d_ws, size_t ws_size,
                              hipStream_t stream) {
    const float* x  = (const float*)d_in[0];
    const int*   ei = (const int*)d_in[1];     // [2, E] int32
    const float* W  = (const float*)d_in[2];   // [64, 128]
    const float* b  = (const float*)d_in[3];   // [64]
    float* out = (float*)d_out;

    const int nNodes = in_sizes[0] / D_FEAT;   // 100000
    const int nEdges = in_sizes[1] / 2;        // 1600000
    const int* row = ei;                       // aggregation targets
    const int* col = ei + nEdges;              // gathered neighbours

    float* accum = (float*)d_ws;                        // [nNodes, 64]
    float* cnt   = accum + (size_t)nNodes * D_FEAT;     // [nNodes]

    // 1) zero accumulator + counts (contiguous in ws)
    int zn = nNodes * (D_FEAT + 1);
    sage_zero_kernel<<<(zn + 255) / 256, 256, 0, stream>>>(accum, zn);

    // 2) edge scatter: one wave32 per edge
    long long sthreads = (long long)nEdges * 32;
    int sblocks = (int)((sthreads + 255) / 256);
    sage_scatter_kernel<<<sblocks, 256, 0, stream>>>(x, row, col, accum, cnt, nEdges);

    // 3) fused mean/concat/GEMM/bias/ReLU, 16 nodes per block
    int gblocks = (nNodes + 15) / 16;
    sage_gemm_kernel<<<gblocks, 128, 0, stream>>>(x, accum, cnt, W, b, out, nNodes);
}